// GCN_48954037240468
// MI455X (gfx1250) — compile-verified
//
#include <hip/hip_runtime.h>
#include <hip/hip_bf16.h>

typedef __attribute__((ext_vector_type(2))) float v2f;
typedef __attribute__((ext_vector_type(8))) float v8f;

// ---------------------------------------------------------------- degree prep
__global__ __launch_bounds__(256) void k_fill1(float* __restrict__ p, int n) {
  int i = blockIdx.x * 256 + threadIdx.x;
  if (i < n) p[i] = 1.0f;  // self-loop contributes 1 to every degree
}

__global__ __launch_bounds__(256) void k_degcount(const int* __restrict__ dst,
                                                  float* __restrict__ deg, int E) {
  int i = blockIdx.x * 256 + threadIdx.x;
  if (i < E) unsafeAtomicAdd(&deg[dst[i]], 1.0f);
}

__global__ __launch_bounds__(256) void k_rsqrt_inplace(float* __restrict__ p, int n) {
  int i = blockIdx.x * 256 + threadIdx.x;
  if (i < n) p[i] = rsqrtf(p[i]);  // deg >= 1 always (self loops)
}

// ------------------------------------------------- f32 WMMA GEMM  C[M,N]=A[M,K]*B[K,N]
// One wave computes a full 16-row strip across all N columns: the A fragment
// (16x4 f32, 2 VGPRs) is loaded once per k-step and reused for N/16 WMMAs.
template <int K, int N>
__global__ __launch_bounds__(256) void k_gemm_wmma(const float* __restrict__ A,
                                                   const float* __restrict__ B,
                                                   float* __restrict__ C, int M) {
  constexpr int NT = N / 16;
  const int wave = blockIdx.x * (256 >> 5) + (threadIdx.x >> 5);
  const int m0 = wave * 16;
  if (m0 >= M) return;                 // wave-uniform: EXEC stays all-ones for WMMA
  const int lane = threadIdx.x & 31;
  const int half = lane >> 4;          // 0: K={k,k+1}, 1: K={k+2,k+3}
  const int l16  = lane & 15;

  v8f acc[NT];
#pragma unroll
  for (int t = 0; t < NT; ++t) acc[t] = (v8f)0.0f;

  const float* arow = A + (size_t)(m0 + l16) * K;
  for (int k = 0; k < K; k += 4) {
    const int ka = k + half * 2;
    const v2f a = *(const v2f*)(arow + ka);  // A 16x4 fragment per ISA layout
#pragma unroll
    for (int t = 0; t < NT; ++t) {
      const float* bp = B + (size_t)ka * N + t * 16 + l16;
      v2f b;
      b.x = bp[0];   // row K=ka,   cols n0..n0+15 (coalesced 64B per half-wave)
      b.y = bp[N];   // row K=ka+1
      acc[t] = __builtin_amdgcn_wmma_f32_16x16x4_f32(
          false, a, false, b, (short)0, acc[t], false, false);
    }
  }
  // C/D layout: VGPR r, lanes 0-15 -> M=m0+r, lanes 16-31 -> M=m0+r+8
#pragma unroll
  for (int t = 0; t < NT; ++t)
#pragma unroll
    for (int r = 0; r < 8; ++r)
      C[(size_t)(m0 + r + half * 8) * N + (t * 16 + l16)] = acc[t][r];
}

// ------------------------------------------- bias + self-loop term (out init)
// out[i,f] = b[f] + xw[i,f] * dinv[i]^2   (self edge has norm dinv*dinv)
template <int F>
__global__ __launch_bounds__(256) void k_self_bias(const float* __restrict__ xw,
                                                   const float* __restrict__ dinv,
                                                   const float* __restrict__ bias,
                                                   float* __restrict__ out, int n) {
  int i = blockIdx.x * 256 + threadIdx.x;
  int node = i / F, f = i & (F - 1);
  if (node < n) {
    float di = dinv[node];
    out[i] = bias[f] + xw[i] * di * di;
  }
}

// ------------------------------------------------- edge scatter (one wave/edge)
template <int F>
__global__ __launch_bounds__(256) void k_scatter(const int* __restrict__ src,
                                                 const int* __restrict__ dst,
                                                 const float* __restrict__ dinv,
                                                 const float* __restrict__ msg,
                                                 float* __restrict__ out, int E) {
  const int e = blockIdx.x * (256 >> 5) + (threadIdx.x >> 5);
  if (e >= E) return;
  const int lane = threadIdx.x & 31;
  const int s = src[e], d = dst[e];
  const float norm = dinv[s] * dinv[d];
  constexpr int V = F / 32;
  const float* mp = msg + (size_t)s * F + lane * V;
  float* op = out + (size_t)d * F + lane * V;
  float v[V];
  if constexpr (V == 4) {
    const float4 t = *(const float4*)mp;
    v[0] = t.x; v[1] = t.y; v[2] = t.z; v[3] = t.w;
  } else {
    const float2 t = *(const float2*)mp;
    v[0] = t.x; v[1] = t.y;
  }
#pragma unroll
  for (int j = 0; j < V; ++j) unsafeAtomicAdd(&op[j], v[j] * norm);
}

__global__ __launch_bounds__(256) void k_relu(float* __restrict__ p, int n) {
  int i = blockIdx.x * 256 + threadIdx.x;
  if (i < n) p[i] = fmaxf(p[i], 0.0f);
}

// ----------------------------------------------- log_softmax over 64 (wave/node)
__global__ __launch_bounds__(256) void k_logsoftmax64(float* __restrict__ out, int n) {
  const int node = blockIdx.x * (256 >> 5) + (threadIdx.x >> 5);
  if (node >= n) return;
  const int lane = threadIdx.x & 31;
  float* p = out + (size_t)node * 64 + lane * 2;
  float2 v = *(float2*)p;
  float m = fmaxf(v.x, v.y);
#pragma unroll
  for (int off = 16; off; off >>= 1) m = fmaxf(m, __shfl_xor(m, off, 32));
  float s = expf(v.x - m) + expf(v.y - m);
#pragma unroll
  for (int off = 16; off; off >>= 1) s += __shfl_xor(s, off, 32);
  const float l = m + logf(s);
  v.x -= l; v.y -= l;
  *(float2*)p = v;
}

// ---------------------------------------------------------------------- launch
extern "C" void kernel_launch(void* const* d_in, const int* in_sizes, int n_in,
                              void* d_out, int out_size, void* d_ws, size_t ws_size,
                              hipStream_t stream) {
  const float* x  = (const float*)d_in[0];
  const float* W1 = (const float*)d_in[1];
  const float* b1 = (const float*)d_in[2];
  const float* W2 = (const float*)d_in[3];
  const float* b2 = (const float*)d_in[4];
  const int*   ei = (const int*)d_in[5];

  const int n = in_sizes[0] / 512;  // 100000 nodes
  const int E = in_sizes[5] / 2;    // 3.2M edges
  const int* src = ei;
  const int* dst = ei + E;

  // Workspace layout: dinv | xw1 | h1 | hw2   (~128.5 MB total)
  float* dinv = (float*)d_ws;
  float* xw1  = (float*)((char*)d_ws + ((size_t)1 << 19));
  float* h1   = xw1 + (size_t)n * 128;
  float* hw2  = h1  + (size_t)n * 128;
  float* out  = (float*)d_out;

  const int T = 256;
  const int tiles = (n + 15) / 16;           // 6250 M-tiles (exact)

  // degrees -> dinv (in place)
  k_fill1<<<(n + T - 1) / T, T, 0, stream>>>(dinv, n);
  k_degcount<<<(E + T - 1) / T, T, 0, stream>>>(dst, dinv, E);
  k_rsqrt_inplace<<<(n + T - 1) / T, T, 0, stream>>>(dinv, n);

  // layer 1: xw1 = x@W1 ; h1 = b1 + self + scatter ; relu
  k_gemm_wmma<512, 128><<<(tiles + 7) / 8, T, 0, stream>>>(x, W1, xw1, n);
  k_self_bias<128><<<((size_t)n * 128 + T - 1) / T, T, 0, stream>>>(xw1, dinv, b1, h1, n);
  k_scatter<128><<<(E + 7) / 8, T, 0, stream>>>(src, dst, dinv, xw1, h1, E);
  k_relu<<<((size_t)n * 128 + T - 1) / T, T, 0, stream>>>(h1, n * 128);

  // layer 2: hw2 = h1@W2 ; out = b2 + self + scatter ; log_softmax
  k_gemm_wmma<128, 64><<<(tiles + 7) / 8, T, 0, stream>>>(h1, W2, hw2, n);
  k_self_bias<64><<<((size_t)n * 64 + T - 1) / T, T, 0, stream>>>(hw2, dinv, b2, out, n);
  k_scatter<64><<<(E + 7) / 8, T, 0, stream>>>(src, dst, dinv, hw2, out, E);
  k_logsoftmax64<<<(n + 7) / 8, T, 0, stream>>>(out, n);
}